// GraphConstructor_59579786330161
// MI455X (gfx1250) — compile-verified
//
#include <hip/hip_runtime.h>

#define NROWS 8192
#define DFEAT 128
#define TOPK  8
#define EPSF  1e-8f

typedef __attribute__((ext_vector_type(2))) float v2f;
typedef __attribute__((ext_vector_type(8))) float v8f;

// ---------------------------------------------------------------------------
// Kernel 1: one workgroup per row of adj.
// Fused: abs -> min/max -> A_norm (+smooth-loss partial) -> top-8 -> A_sparse/phi.
// All three 256MB outputs written with non-temporal stores (pure streaming).
// ---------------------------------------------------------------------------
__global__ __launch_bounds__(256) void row_kernel(
    const float* __restrict__ adj,
    float* __restrict__ A_norm_out,
    float* __restrict__ A_sparse_out,
    float* __restrict__ phi_out,
    float* __restrict__ tkv,        // [NROWS][TOPK] top-k values
    int*   __restrict__ tki,        // [NROWS][TOPK] top-k indices
    float* __restrict__ ws_smooth)  // [NROWS] per-row smooth-loss partial
{
    __shared__ float a[NROWS];      // 32 KB: row cache (abs, then normalized)
    __shared__ float red[256];
    __shared__ int   redi[256];
    __shared__ float tvs[TOPK];
    __shared__ int   tis[TOPK];
    __shared__ float sMin, sMax;

    const int tid = threadIdx.x;
    const int row = blockIdx.x;
    const float* src = adj + (size_t)row * NROWS;

    // ---- Phase A: load |adj| row, local min/max -------------------------
    float lmax = -__builtin_inff();
    float lmin =  __builtin_inff();
    #pragma unroll 4
    for (int s = 0; s < NROWS / 256; ++s) {
        int col = s * 256 + tid;                       // coalesced
        float v = fabsf(__builtin_nontemporal_load(src + col));
        a[col] = v;
        lmax = fmaxf(lmax, v);
        lmin = fminf(lmin, v);
    }
    red[tid] = lmax; __syncthreads();
    for (int off = 128; off > 0; off >>= 1) {
        if (tid < off) red[tid] = fmaxf(red[tid], red[tid + off]);
        __syncthreads();
    }
    if (tid == 0) sMax = red[0];
    __syncthreads();
    red[tid] = lmin; __syncthreads();
    for (int off = 128; off > 0; off >>= 1) {
        if (tid < off) red[tid] = fminf(red[tid], red[tid + off]);
        __syncthreads();
    }
    if (tid == 0) sMin = red[0];
    __syncthreads();

    const float amin = sMin;
    const float amax = sMax;
    const float inv1 = 1.0f / (amax - amin + EPSF);   // A_norm scale
    const float inv2 = 1.0f / (amax + EPSF);          // A_prev scale
    const float dinv = inv1 - inv2;

    // ---- Phase B: normalize, write A_norm, smooth-loss partial ----------
    float sm = 0.0f;
    float* an_out = A_norm_out + (size_t)row * NROWS;
    #pragma unroll 4
    for (int s = 0; s < NROWS / 256; ++s) {
        int col = s * 256 + tid;
        float at = a[col];
        float an = (at - amin) * inv1;
        float df = (at - amin) * dinv;                // an - a_prev
        sm += df * df;
        a[col] = an;                                  // keep normalized row for top-k
        __builtin_nontemporal_store(an, an_out + col);
    }
    red[tid] = sm; __syncthreads();
    for (int off = 128; off > 0; off >>= 1) {
        if (tid < off) red[tid] += red[tid + off];
        __syncthreads();
    }
    if (tid == 0) ws_smooth[row] = red[0];
    __syncthreads();

    // ---- Phase C: iterative top-8 argmax over LDS row -------------------
    for (int j = 0; j < TOPK; ++j) {
        float bv = -__builtin_inff();
        int   bi = -1;
        #pragma unroll 4
        for (int s = 0; s < NROWS / 256; ++s) {
            int col = s * 256 + tid;
            float v = a[col];
            if (v > bv || (v == bv && col < bi)) { bv = v; bi = col; }
        }
        red[tid] = bv; redi[tid] = bi; __syncthreads();
        for (int off = 128; off > 0; off >>= 1) {
            if (tid < off) {
                float ov = red[tid + off]; int oi = redi[tid + off];
                if (ov > red[tid] || (ov == red[tid] && oi < redi[tid])) {
                    red[tid] = ov; redi[tid] = oi;
                }
            }
            __syncthreads();
        }
        if (tid == 0) {
            tvs[j] = red[0];
            tis[j] = redi[0];
            a[redi[0]] = -1.0f;                       // mask (A_norm >= 0)
        }
        __syncthreads();
    }

    // ---- Phase D: write A_sparse and phi (phi == A_sparse, nonneg) ------
    float tv_[TOPK]; int ti_[TOPK];
    #pragma unroll
    for (int j = 0; j < TOPK; ++j) { tv_[j] = tvs[j]; ti_[j] = tis[j]; }

    float* as_out = A_sparse_out + (size_t)row * NROWS;
    float* ph_out = phi_out      + (size_t)row * NROWS;
    #pragma unroll 4
    for (int s = 0; s < NROWS / 256; ++s) {
        int col = s * 256 + tid;
        float v = 0.0f;
        #pragma unroll
        for (int j = 0; j < TOPK; ++j)
            if (col == ti_[j]) v = tv_[j];
        __builtin_nontemporal_store(v, as_out + col);
        __builtin_nontemporal_store(v, ph_out + col);
    }

    if (tid < TOPK) {
        tkv[row * TOPK + tid] = tvs[tid];
        tki[row * TOPK + tid] = tis[tid];
    }
}

// ---------------------------------------------------------------------------
// Kernel 2: Z_hat = A_sparse @ Z_t via WMMA on the compacted form.
// Per 16-row block: A_mini[16x128] is block-diagonal (row r nonzero at cols
// 8r..8r+7 = its top-8 values), B_mini[128x128] gathers the referenced Z_t
// rows. Exact product. 8 waves; wave w owns N-tile w*16; K=128 via
// v_wmma_f32_16x16x4_f32, B staged in one LDS buffer in two 64-row phases.
// ---------------------------------------------------------------------------
__global__ __launch_bounds__(256) void zhat_wmma_kernel(
    const float* __restrict__ Zt,
    const float* __restrict__ tkv,
    const int*   __restrict__ tki,
    float* __restrict__ Zhat,
    float* __restrict__ ws_recon)   // [NROWS/16] per-block recon partial
{
    __shared__ float Am[16][132];     // padded: lane-strided K-reads hit distinct banks
    __shared__ float Bm[64][DFEAT];   // one 64-row phase of B_mini
    __shared__ int   srcs[128];
    __shared__ float red[256];

    const int tid  = threadIdx.x;
    const int blk  = blockIdx.x;      // 512 blocks
    const int row0 = blk * 16;

    for (int s = tid; s < 16 * 132; s += 256) (&Am[0][0])[s] = 0.0f;
    __syncthreads();
    if (tid < 128) {
        int r = tid >> 3, j = tid & 7;
        float v = tkv[(row0 + r) * TOPK + j];
        int  ix = tki[(row0 + r) * TOPK + j];
        Am[r][r * 8 + j] = v;         // block-diagonal compacted A
        srcs[tid] = ix;
    }
    __syncthreads();

    const int wave = tid >> 5;
    const int lane = tid & 31;
    const int n0   = wave * 16;
    const int m    = lane & 15;
    const int ksel = (lane < 16) ? 0 : 2;

    v8f c = {};
    for (int p = 0; p < 2; ++p) {
        __syncthreads();              // previous phase's B fully consumed
        {   // gather 64 rows of Z_t: 4 threads/row, 8 x float4 each
            int r    = tid >> 2;
            int part = (tid & 3) * 32;
            const float4* sp = (const float4*)(Zt + (size_t)srcs[p * 64 + r] * DFEAT + part);
            float4* dp = (float4*)(&Bm[r][part]);
            #pragma unroll
            for (int q = 0; q < 8; ++q) dp[q] = sp[q];
        }
        __syncthreads();

        #pragma unroll
        for (int k0 = 0; k0 < 64; k0 += 4) {
            int kg = p * 64 + k0;     // global K index for A
            v2f av, bv;
            av.x = Am[m][kg + ksel];
            av.y = Am[m][kg + ksel + 1];
            bv.x = Bm[k0 + ksel][n0 + m];
            bv.y = Bm[k0 + ksel + 1][n0 + m];
            c = __builtin_amdgcn_wmma_f32_16x16x4_f32(
                    false, av, false, bv, (short)0, c, false, false);
        }
    }

    // C layout: VGPR v -> row v (lanes 0-15) / row v+8 (lanes 16-31), col = m
    float rc = 0.0f;
    const int mbase = (lane < 16) ? 0 : 8;
    #pragma unroll
    for (int v = 0; v < 8; ++v) {
        int r   = row0 + mbase + v;
        int col = n0 + m;
        float zh = c[v];
        __builtin_nontemporal_store(zh, Zhat + (size_t)r * DFEAT + col);
        float d = zh - Zt[(size_t)r * DFEAT + col];
        rc += d * d;
    }
    red[tid] = rc; __syncthreads();
    for (int off = 128; off > 0; off >>= 1) {
        if (tid < off) red[tid] += red[tid + off];
        __syncthreads();
    }
    if (tid == 0) ws_recon[blk] = red[0];
}

// ---------------------------------------------------------------------------
// Kernel 3: deterministic final reductions for the two loss scalars.
// ---------------------------------------------------------------------------
__global__ __launch_bounds__(256) void finalize_kernel(
    const float* __restrict__ ws_smooth,
    const float* __restrict__ ws_recon,
    float* __restrict__ Lrec,
    float* __restrict__ Lsm)
{
    __shared__ float red[256];
    const int tid = threadIdx.x;

    float s = 0.0f;
    for (int i = tid; i < NROWS; i += 256) s += ws_smooth[i];
    red[tid] = s; __syncthreads();
    for (int off = 128; off > 0; off >>= 1) {
        if (tid < off) red[tid] += red[tid + off];
        __syncthreads();
    }
    if (tid == 0) *Lsm = red[0] / ((float)NROWS * (float)NROWS);
    __syncthreads();

    float r = 0.0f;
    for (int i = tid; i < NROWS / 16; i += 256) r += ws_recon[i];
    red[tid] = r; __syncthreads();
    for (int off = 128; off > 0; off >>= 1) {
        if (tid < off) red[tid] += red[tid + off];
        __syncthreads();
    }
    if (tid == 0) *Lrec = red[0] / ((float)NROWS * (float)DFEAT);
}

// ---------------------------------------------------------------------------
extern "C" void kernel_launch(void* const* d_in, const int* in_sizes, int n_in,
                              void* d_out, int out_size, void* d_ws, size_t ws_size,
                              hipStream_t stream)
{
    const float* Zt  = (const float*)d_in[0];   // [8192,128]
    // d_in[1] = Z_prev: unused by the reference math
    const float* adj = (const float*)d_in[2];   // [8192,8192]

    float* out = (float*)d_out;
    const size_t NN = (size_t)NROWS * NROWS;

    float* Zhat    = out;                                   // 8192*128
    float* Anorm   = out + (size_t)NROWS * DFEAT;           // 8192*8192
    float* Asparse = Anorm + NN;                            // 8192*8192
    float* phi     = Asparse + NN;                          // 8192*8192
    float* Lrec    = phi + NN;                              // scalar
    float* Lsm     = Lrec + 1;                              // scalar

    float* tkv       = (float*)d_ws;                        // 8192*8 f32
    int*   tki       = (int*)(tkv + (size_t)NROWS * TOPK);  // 8192*8 i32
    float* ws_smooth = (float*)(tki + (size_t)NROWS * TOPK);// 8192 f32
    float* ws_recon  = ws_smooth + NROWS;                   // 512 f32

    row_kernel<<<NROWS, 256, 0, stream>>>(adj, Anorm, Asparse, phi,
                                          tkv, tki, ws_smooth);
    zhat_wmma_kernel<<<NROWS / 16, 256, 0, stream>>>(Zt, tkv, tki, Zhat, ws_recon);
    finalize_kernel<<<1, 256, 0, stream>>>(ws_smooth, ws_recon, Lrec, Lsm);
}